// HiPPO_LegT_36094905155670
// MI455X (gfx1250) — compile-verified
//
#include <hip/hip_runtime.h>
#include <hip/hip_bf16.h>

typedef __attribute__((ext_vector_type(2))) float v2f;
typedef __attribute__((ext_vector_type(8))) float v8f;

#define NST    128
#define LEN    2048
#define NBATCH 512
#define KPAD   80         // zero pad in front of K in LDS so K[d<0] reads 0
#define BMB    64         // block batch tile
#define BTB    64         // block time tile

// ---------------------------------------------------------------------------
// Stage 1: sequential recurrence w_{k+1} = A w_k, w_0 = C_discr.
// One workgroup, 1024 threads = 8 threads per state row. A rows live in
// registers (16 floats/thread), w double-buffered in LDS, one barrier/step.
// w_k goes to workspace so K[k] = B . w_k is computed massively parallel
// afterwards (keeps the critical chain to 1 barrier per step).
// ---------------------------------------------------------------------------
__global__ __launch_bounds__(1024) void hippo_wseq_kernel(
    const float* __restrict__ A, const float* __restrict__ C,
    float* __restrict__ Wall)
{
    __shared__ float wbuf[2][NST];
    const int tid = threadIdx.x;
    const int r = tid >> 3;   // state row 0..127
    const int p = tid & 7;    // partial id 0..7

    float ra[16];
#pragma unroll
    for (int j = 0; j < 16; ++j) ra[j] = A[r * NST + p * 16 + j];

    if (tid < NST) wbuf[0][tid] = C[tid];
    __syncthreads();

    for (int k = 0; k < LEN; ++k) {
        const float* w = wbuf[k & 1];
        if (p == 0) Wall[k * NST + r] = w[r];   // w_k for later B-dot
        float s = 0.f;
#pragma unroll
        for (int j = 0; j < 16; ++j) s += ra[j] * w[p * 16 + j];
        // reduce the 8 partials (aligned group inside one wave32)
        s += __shfl_xor(s, 1, 32);
        s += __shfl_xor(s, 2, 32);
        s += __shfl_xor(s, 4, 32);
        if (p == 0) wbuf[(k & 1) ^ 1][r] = s;
        __syncthreads();
    }
}

// ---------------------------------------------------------------------------
// Stage 1b: K[k] = dot(B, w_k). 2048 independent 128-length dots.
// ---------------------------------------------------------------------------
__global__ __launch_bounds__(256) void hippo_kdot_kernel(
    const float* __restrict__ Wall, const float* __restrict__ B,
    float* __restrict__ K)
{
    const int k = blockIdx.x * blockDim.x + threadIdx.x;
    if (k >= LEN) return;
    const float* wr = Wall + k * NST;
    float s = 0.f;
#pragma unroll 4
    for (int j = 0; j < NST; ++j) s += B[j] * wr[j];
    K[k] = s;
}

// ---------------------------------------------------------------------------
// Stage 2: triangular-Toeplitz "convolution as GEMM" with fp32 WMMA.
//   P[b,t] = f_t * sum_{s<=t} U[b,s] * K[t-s] + D0 * U[b,t]
// Block: 256 threads (8 waves), tile 64 (batch) x 64 (time). Each wave owns
// TWO 16x16 output tiles stacked in batch (rows +0 and +16), so each B
// (kernel) fragment is reused by two WMMAs -> 2 ds ops per wmma.
// Time-tile index is reversed so the heaviest (most-reduction) blocks launch
// first, balancing the triangular workload.
// ---------------------------------------------------------------------------
__global__ __launch_bounds__(256) void hippo_conv_wmma_kernel(
    const float* __restrict__ U, const float* __restrict__ Kg,
    const float* __restrict__ Dg, float* __restrict__ out)
{
    __shared__ float Klds[KPAD + LEN];    // zero-padded kernel (8.5 KB)
    __shared__ float Ulds[BMB][68];       // 64x64 chunk, stride 68 (16B align, conflict-free)

    const int tid  = threadIdx.x;
    const int lane = tid & 31;
    const int wid  = tid >> 5;

    const int t0_blk = ((int)gridDim.x - 1 - (int)blockIdx.x) * BTB; // heavy blocks first
    const int b0_blk = blockIdx.y * BMB;

    // stage full K into LDS with leading zeros (branch-free K[d<0] = 0)
    for (int i = tid; i < KPAD; i += 256) Klds[i] = 0.f;
    for (int i = tid; i < LEN; i += 256) Klds[KPAD + i] = Kg[i];

    const int bp    = (wid & 1) * 32;          // wave's batch pair: rows bp+m, bp+16+m
    const int t0    = t0_blk + (wid >> 1) * 16; // wave's time tile
    const int half  = lane >> 4;               // ISA A/B fragment half-select
    const int m     = lane & 15;               // row (A) / col (B) within tile

    v8f acc0 = {};
    v8f acc1 = {};

    const int s_end = t0_blk + BTB;            // only s <= t_blockmax contributes
    for (int s0 = 0; s0 < s_end; s0 += 64) {
        // cooperative coalesced load of U[b0..b0+63, s0..s0+63] (16 floats/thread)
        {
            const int row = tid >> 2;
            const int col = (tid & 3) * 16;
            const float* src = U + (size_t)(b0_blk + row) * LEN + s0 + col;
#pragma unroll
            for (int q = 0; q < 4; ++q)
                *(float4*)&Ulds[row][col + 4 * q] = *(const float4*)(src + 4 * q);
            if (s0 + 64 < s_end) __builtin_prefetch(src + 64, 0, 3);
        }
        __syncthreads();

        if (s0 <= t0 + 15) {                   // wave-uniform skip (EXEC stays all-1)
#pragma unroll
            for (int s = 0; s < 64; s += 4) {
                // A fragments (16x4 f32): lanes 0-15 -> k=0,1 ; lanes 16-31 -> k=2,3
                v2f a0 = *(const v2f*)&Ulds[bp + m][s + 2 * half];
                v2f a1 = *(const v2f*)&Ulds[bp + 16 + m][s + 2 * half];
                // B fragment (4x16 f32): M[s_row, t_col] = K[t - s_row], shared by both
                const int d0 = t0 + m - (s0 + s) - 2 * half;
                v2f b;
                b.x = Klds[KPAD + d0];
                b.y = Klds[KPAD + d0 - 1];
                acc0 = __builtin_amdgcn_wmma_f32_16x16x4_f32(
                    false, a0, false, b, (short)0, acc0, false, false);
                acc1 = __builtin_amdgcn_wmma_f32_16x16x4_f32(
                    false, a1, false, b, (short)0, acc1, false, false);
            }
        }
        __syncthreads();
    }

    // epilogue: P = f * acc + D0 * u   (C/D layout: VGPR v -> rows v / v+8)
    const float D0 = Dg[0];
    const int col  = t0 + m;
    const float f  = (col == 0) ? 1.f : 2.f;
#pragma unroll
    for (int v = 0; v < 8; ++v) {
        const int rowa = b0_blk + bp + half * 8 + v;
        const int rowb = rowa + 16;
        const float ua = U[(size_t)rowa * LEN + col];
        const float ub = U[(size_t)rowb * LEN + col];
        out[(size_t)rowa * LEN + col] = f * acc0[v] + D0 * ua;
        out[(size_t)rowb * LEN + col] = f * acc1[v] + D0 * ub;
    }
}

// ---------------------------------------------------------------------------
extern "C" void kernel_launch(void* const* d_in, const int* in_sizes, int n_in,
                              void* d_out, int out_size, void* d_ws, size_t ws_size,
                              hipStream_t stream) {
    (void)in_sizes; (void)n_in; (void)out_size; (void)ws_size;
    const float* U  = (const float*)d_in[0];   // (512, 2048)
    const float* A  = (const float*)d_in[1];   // (128, 128)
    const float* Bv = (const float*)d_in[2];   // (128,)
    const float* Cd = (const float*)d_in[3];   // (128,)
    const float* Dd = (const float*)d_in[4];   // (1,)
    float* out = (float*)d_out;                // (512, 2048)

    float* Wall = (float*)d_ws;                // 2048*128 floats = 1 MB
    float* Kg   = Wall + LEN * NST;            // 2048 floats

    hipLaunchKernelGGL(hippo_wseq_kernel, dim3(1), dim3(1024), 0, stream, A, Cd, Wall);
    hipLaunchKernelGGL(hippo_kdot_kernel, dim3(LEN / 256), dim3(256), 0, stream, Wall, Bv, Kg);
    hipLaunchKernelGGL(hippo_conv_wmma_kernel, dim3(LEN / BTB, NBATCH / BMB), dim3(256),
                       0, stream, U, Kg, Dd, out);
}